// Patcher_43705587204236
// MI455X (gfx1250) — compile-verified
//
#include <hip/hip_runtime.h>
#include <hip/hip_bf16.h>
#include <stdint.h>

#define MS   32   // MAX_SPACE_F
#define MT   1    // MAX_TIME_F
#define NREG 10   // NUM_REGIONS

#define AS1 __attribute__((address_space(1)))
#define AS3 __attribute__((address_space(3)))

typedef int v4i __attribute__((vector_size(16)));

#if __has_builtin(__builtin_amdgcn_global_load_async_to_lds_b128) && \
    __has_builtin(__builtin_amdgcn_global_store_async_from_lds_b128)
#define USE_ASYNC_LDS 1
#else
#define USE_ASYNC_LDS 0
#endif

__device__ __forceinline__ void wait_asynccnt0() {
#if __has_builtin(__builtin_amdgcn_s_wait_asynccnt)
  __builtin_amdgcn_s_wait_asynccnt(0);
#else
  asm volatile("s_wait_asynccnt 0" ::: "memory");
#endif
}

// ---------------------------------------------------------------------------
// 1) Bulk copy spikes -> patches (identity under mt=1, ms=32).
//    CDNA5 path: async DMA global -> LDS -> global (no VGPR data movement),
//    512 B per wave per b128 op, tracked with ASYNCcnt.
// ---------------------------------------------------------------------------
__global__ void k_patch_copy(const float4* __restrict__ src,
                             float4* __restrict__ dst, long long n4) {
#if USE_ASYNC_LDS
  __shared__ float4 stage[8][32];                 // one 512B slot per wave
  const int wave = threadIdx.x >> 5;
  const int lane = threadIdx.x & 31;
  const long long gw    = (long long)blockIdx.x * (blockDim.x >> 5) + wave;
  const long long wstep = (long long)gridDim.x * (blockDim.x >> 5);
  AS3 v4i* lp = (AS3 v4i*)&stage[wave][lane];
  for (long long base = gw * 32; base < n4; base += wstep * 32) {
    long long i = base + lane;                    // loop bound uniform per wave
    if (i < n4) {
      __builtin_amdgcn_global_load_async_to_lds_b128(
          (AS1 v4i*)(uintptr_t)(src + i), lp, 0, 0);
      wait_asynccnt0();
      __builtin_amdgcn_global_store_async_from_lds_b128(
          (AS1 v4i*)(uintptr_t)(dst + i), lp, 0, 0);
      wait_asynccnt0();                           // LDS slot reusable next iter
    }
  }
#else
  long long i    = (long long)blockIdx.x * blockDim.x + threadIdx.x;
  long long step = (long long)gridDim.x * blockDim.x;
  for (; i < n4; i += step) {
    __builtin_prefetch(src + i + step, 0, 0);     // global_prefetch_b8
    dst[i] = src[i];
  }
#endif
}

__global__ void k_tail_copy(const float* __restrict__ src,
                            float* __restrict__ dst,
                            long long start, int rem) {
  int t = threadIdx.x;
  if (t < rem) dst[start + t] = src[start + t];
}

// ---------------------------------------------------------------------------
// 2) Pad fix-up: patch s=pad_idx gets fixed[c] = c<pad_rem ? cur[c] : prev[c].
//    pad_space_len lives on-device (graph capture: no host readback).
// ---------------------------------------------------------------------------
__global__ void k_fixup(const float* __restrict__ spikes,
                        float* __restrict__ out,
                        const int* __restrict__ pad_space_len,
                        int T, int N, int nT, long long total) {
  const int psl     = pad_space_len[0];
  const int pad_rem = psl % MS;
  if (pad_rem == 0) return;                       // reference: no fix
  const int pad_idx = psl / MS;
  if (pad_idx <= 0) return;

  long long idx = (long long)blockIdx.x * blockDim.x + threadIdx.x;
  if (idx >= total) return;                       // total = B*nT*C, C = MT*MS
  const int c  = (int)(idx % (MT * MS));
  long long bt = idx / (MT * MS);
  const int t  = (int)(bt % nT);
  const long long b = bt / nT;

  const long long row = (b * T + t) * (long long)N;   // mt==1 flat mapping
  const int s = (c < pad_rem) ? pad_idx : pad_idx - 1;
  out[row + (long long)pad_idx * MS + c] = spikes[row + (long long)s * MS + c];
}

// ---------------------------------------------------------------------------
// 3) Tiny reductions into workspace: t_any[nT], s_any[nS], mode[nS].
// ---------------------------------------------------------------------------
__global__ void k_reduce(const int* __restrict__ tmask,
                         const int* __restrict__ smask,
                         const int* __restrict__ region,
                         int* __restrict__ ws,
                         int B, int T, int N, int nT, int nS) {
  int* t_any = ws;
  int* s_any = ws + nT;
  int* modep = ws + nT + nS;

  for (int t = threadIdx.x; t < nT; t += blockDim.x) {
    int any = 0;
    for (int b = 0; b < B; ++b)
      for (int j = 0; j < MT; ++j)
        any |= (tmask[(long long)b * T + t * MT + j] == 1);
    t_any[t] = any;
  }
  for (int s = threadIdx.x; s < nS; s += blockDim.x) {
    int any = 0;
    for (int b = 0; b < B; ++b)
      for (int j = 0; j < MS; ++j)
        any |= (smask[(long long)b * N + s * MS + j] == 1);
    s_any[s] = any;

    int counts[NREG];
#pragma unroll
    for (int r = 0; r < NREG; ++r) counts[r] = 0;
    for (int j = 0; j < MS; ++j) {
      int r = region[s * MS + j];                 // batch 0 only (reference)
      if (r >= 0 && r < NREG) counts[r]++;
    }
    int best = 0, bestc = counts[0];
#pragma unroll
    for (int r = 1; r < NREG; ++r)
      if (counts[r] > bestc) { bestc = counts[r]; best = r; }  // first-max wins
    modep[s] = best;
  }
}

// ---------------------------------------------------------------------------
// 4) Broadcast 5 outputs (written as float values; tuple sections follow
//    patches at offset P, each of length Q = B*nT*nS).
// ---------------------------------------------------------------------------
__global__ void k_broadcast(float* __restrict__ out,
                            const int* __restrict__ ws,
                            long long P, long long Q, int nT, int nS) {
  const int* t_any = ws;
  const int* s_any = ws + nT;
  const int* modep = ws + nT + nS;

  long long idx = (long long)blockIdx.x * blockDim.x + threadIdx.x;
  if (idx >= Q) return;
  const long long p = idx % ((long long)nT * nS);
  const int t = (int)(p / nS);
  const int s = (int)(p % nS);

  out[P + 0 * Q + idx] = s_any[s] ? 1.0f : 0.0f;  // space_out
  out[P + 1 * Q + idx] = t_any[t] ? 1.0f : 0.0f;  // time_out
  out[P + 2 * Q + idx] = (float)s;                // spacestamps
  out[P + 3 * Q + idx] = (float)t;                // timestamps
  out[P + 4 * Q + idx] = (float)modep[s];         // regionstamps
}

// ---------------------------------------------------------------------------
extern "C" void kernel_launch(void* const* d_in, const int* in_sizes, int n_in,
                              void* d_out, int out_size, void* d_ws, size_t ws_size,
                              hipStream_t stream) {
  const float* spikes = (const float*)d_in[0];
  const int* pad_space = (const int*)d_in[1];
  // d_in[2] = pad_time_len (unused by the reference outputs)
  const int* tmask  = (const int*)d_in[3];
  const int* smask  = (const int*)d_in[4];
  const int* region = (const int*)d_in[5];
  float* out = (float*)d_out;

  const long long total = in_sizes[0];   // B*T*N
  const long long bt    = in_sizes[3];   // B*T
  const long long bn    = in_sizes[4];   // B*N
  const long long B = (bt * bn) / total;
  const long long T = bt / B;
  const long long N = bn / B;
  const int nT = (int)((T + MT - 1) / MT);
  const int nS = (int)((N + MS - 1) / MS);
  const long long P = total;             // patches element count
  const long long Q = B * (long long)nT * nS;

  // 1) bulk identity copy (float4 granularity)
  const long long n4  = total / 4;
  const int rem       = (int)(total & 3);
  long long cb        = (n4 + 255) / 256;
  int copy_blocks     = (int)(cb > 32768 ? 32768 : cb);
  if (copy_blocks < 1) copy_blocks = 1;
  k_patch_copy<<<copy_blocks, 256, 0, stream>>>(
      (const float4*)spikes, (float4*)out, n4);
  if (rem) k_tail_copy<<<1, 4, 0, stream>>>(spikes, out, n4 * 4, rem);

  // 2) pad fix-up (depends on stream order after the copy)
  const long long fixTot = B * (long long)nT * (MT * MS);
  int fix_blocks = (int)((fixTot + 255) / 256);
  k_fixup<<<fix_blocks, 256, 0, stream>>>(spikes, out, pad_space,
                                          (int)T, (int)N, nT, fixTot);

  // 3) tiny reductions into workspace ((nT + 2*nS) ints)
  k_reduce<<<1, 256, 0, stream>>>(tmask, smask, region, (int*)d_ws,
                                  (int)B, (int)T, (int)N, nT, nS);

  // 4) broadcast the 5 scalar-per-token outputs
  int bc_blocks = (int)((Q + 255) / 256);
  k_broadcast<<<bc_blocks, 256, 0, stream>>>(out, (const int*)d_ws, P, Q, nT, nS);
}